// Heads_WTA_17532056502512
// MI455X (gfx1250) — compile-verified
//
#include <hip/hip_runtime.h>
#include <math.h>

// Heads_WTA: per-row masked top-2 of 8, weighted by softmax(W) + mean of top-2.
// Memory-bound streaming kernel: 44 B/row, 369 MB total, ~16 us at 23.3 TB/s.
// No matmul structure -> no WMMA (intensity ~0.7 FLOP/B, far below matrix ridge).

typedef float        v4f __attribute__((ext_vector_type(4)));
typedef unsigned int v2u __attribute__((ext_vector_type(2)));

#define NM 8

__device__ __forceinline__ float bcast_f32(float v) {
    // block-uniform value -> SGPR via readfirstlane
    union { float f; unsigned u; } in, out;
    in.f = v;
    out.u = __builtin_amdgcn_readfirstlane(in.u);
    return out.f;
}

__global__ __launch_bounds__(256)
void wta_topk2_kernel(const float* __restrict__ x,
                      const unsigned char* __restrict__ mask,
                      const float* __restrict__ W,
                      float* __restrict__ out,
                      int nrows)
{
    // ---- per-block softmax(W) into LDS (8 values, trivial) ----
    __shared__ float s_sw[NM];
    if (threadIdx.x == 0) {
        float w[NM];
        float m = -INFINITY;
        #pragma unroll
        for (int j = 0; j < NM; ++j) { w[j] = W[j]; m = fmaxf(m, w[j]); }
        float s = 0.0f;
        #pragma unroll
        for (int j = 0; j < NM; ++j) { w[j] = expf(w[j] - m); s += w[j]; }
        float inv = 1.0f / s;
        #pragma unroll
        for (int j = 0; j < NM; ++j) s_sw[j] = w[j] * inv;
    }
    __syncthreads();

    // broadcast weights to SGPRs (uniform across the block)
    float sw[NM];
    #pragma unroll
    for (int j = 0; j < NM; ++j) sw[j] = bcast_f32(s_sw[j]);

    int row = blockIdx.x * blockDim.x + threadIdx.x;
    if (row >= nrows) return;

    // ---- streaming loads: 2x b128 (x) + 1x b64 (mask bytes), all TH_NT ----
    const v4f* x4 = (const v4f*)(x + (size_t)row * NM);
    v4f a = __builtin_nontemporal_load(x4);
    v4f b = __builtin_nontemporal_load(x4 + 1);
    v2u mv = __builtin_nontemporal_load((const v2u*)(mask + (size_t)row * NM));

    float xv[NM] = { a.x, a.y, a.z, a.w, b.x, b.y, b.z, b.w };

    // ---- masked top-2 scan carrying (value, weight) pairs ----
    // strict '>' => first occurrence wins on ties, matching jax.lax.top_k
    float v1 = -INFINITY, v2 = -INFINITY;
    float w1 = 0.0f, w2 = 0.0f;
    #pragma unroll
    for (int j = 0; j < NM; ++j) {
        unsigned bits = (j < 4) ? mv.x : mv.y;
        bool valid = ((bits >> ((j & 3) * 8)) & 0xFFu) != 0u;
        float val = valid ? xv[j] : -INFINITY;

        bool gt1 = val > v1;
        bool gt2 = val > v2;
        float nv2 = gt1 ? v1 : (gt2 ? val   : v2);
        float nw2 = gt1 ? w1 : (gt2 ? sw[j] : w2);
        v1 = gt1 ? val   : v1;
        w1 = gt1 ? sw[j] : w1;
        v2 = nv2;
        w2 = nw2;
    }

    // out = v1*sw[i1] + v2*sw[i2] + mean(v1, v2)
    float o = fmaf(v1, w1, v2 * w2) + 0.5f * (v1 + v2);
    __builtin_nontemporal_store(o, out + row);
}

extern "C" void kernel_launch(void* const* d_in, const int* in_sizes, int n_in,
                              void* d_out, int out_size, void* d_ws, size_t ws_size,
                              hipStream_t stream) {
    const float*         x    = (const float*)d_in[0];         // [B, 8] f32
    const unsigned char* mask = (const unsigned char*)d_in[1]; // [B, 8] bool (1 B/elem)
    const float*         W    = (const float*)d_in[2];         // [8, 1] f32
    float*               out  = (float*)d_out;                 // [B, 1] f32

    int nrows = in_sizes[0] / NM;
    int threads = 256;                       // 8 wave32 per block
    int blocks = (nrows + threads - 1) / threads;

    wta_topk2_kernel<<<blocks, threads, 0, stream>>>(x, mask, W, out, nrows);
}